// SNNWithETADImproved_43147241456211
// MI455X (gfx1250) — compile-verified
//
#include <hip/hip_runtime.h>
#include <cstdint>
#include <cstddef>

// Problem constants (reference: T, B, D)
#define T_STEPS 20
#define BATCH   1024
#define DFEAT   2312
#define ROWS_PER_WG 4
#define THREADS 128                              // 4 waves (wave32), one wave per row
#define NWG     (BATCH / ROWS_PER_WG)            // 256 workgroups (persistent grid)
#define KMAX    73                               // ceil(2312/32) columns per lane
#define DPAD    (KMAX * 32)                      // 2336 padded row length in LDS
#define NVEC_ROW (DFEAT / 4)                     // 578 x 16B vectors per row (exact)
#define PF_PER_THREAD ((NVEC_ROW + THREADS - 1) / THREADS)   // 5
#define PF_WAVE_ISSUES (ROWS_PER_WG * PF_PER_THREAD)         // 20 async issues per wave/tile

// Types matching the builtin's probed signature: pointer to 16B int vector,
// global (AS1, printed "__device__") for src and LDS (AS3) for dst.
typedef int v4i __attribute__((__vector_size__(16)));
typedef __attribute__((address_space(1))) v4i as1_v4i;
typedef __attribute__((address_space(3))) v4i as3_v4i;
typedef __attribute__((address_space(3))) void as3_void;

// ---- CDNA5 async global->LDS DMA (ASYNCcnt-tracked) ------------------------
__device__ __forceinline__ void async_copy_b128(void* lds_dst, const void* gsrc) {
#if __has_builtin(__builtin_amdgcn_global_load_async_to_lds_b128)
  __builtin_amdgcn_global_load_async_to_lds_b128(
      (as1_v4i*)(uintptr_t)gsrc,      // flat VA == global VA for global memory
      (as3_v4i*)lds_dst,              // explicit generic->LDS addrspacecast
      0, 0);
#else
  unsigned l = (unsigned)(uintptr_t)(as3_void*)lds_dst;
  asm volatile("global_load_async_to_lds_b128 %0, %1, off"
               :: "v"(l), "v"(gsrc) : "memory");
#endif
}

__device__ __forceinline__ void s_wait_asynccnt_prev_tile() {
#if __has_builtin(__builtin_amdgcn_s_wait_asynccnt)
  __builtin_amdgcn_s_wait_asynccnt(PF_WAVE_ISSUES);   // wait: only newest tile outstanding
#else
  asm volatile("s_wait_asynccnt 20" ::: "memory");
#endif
}
__device__ __forceinline__ void s_wait_asynccnt_zero() {
#if __has_builtin(__builtin_amdgcn_s_wait_asynccnt)
  __builtin_amdgcn_s_wait_asynccnt(0);
#else
  asm volatile("s_wait_asynccnt 0" ::: "memory");
#endif
}

// ---- helpers ---------------------------------------------------------------
__device__ __forceinline__ float sigmoidf_fast(float v) {
  return __builtin_amdgcn_rcpf(1.0f + __expf(-v));
}
__device__ __forceinline__ float wave_sum(float v) {
#pragma unroll
  for (int off = 16; off > 0; off >>= 1) v += __shfl_xor(v, off, 32);
  return v;
}
__device__ __forceinline__ float wave_max(float v) {
#pragma unroll
  for (int off = 16; off > 0; off >>= 1) v = fmaxf(v, __shfl_xor(v, off, 32));
  return v;
}

// ---- kernels ---------------------------------------------------------------
__global__ void init_ctl_kernel(unsigned* ctl) {
  if (threadIdx.x == 0 && blockIdx.x == 0) *ctl = 0u;
}

__global__ __launch_bounds__(THREADS, 1)
void snn_atic_lif_persistent(const float* __restrict__ x,
                             const float* __restrict__ ac,
                             const float* __restrict__ tw,
                             const float* __restrict__ ig,
                             const float* __restrict__ p_md,
                             const float* __restrict__ p_sd,
                             const float* __restrict__ p_th,
                             float* __restrict__ out,
                             unsigned* ctl,          // grid-barrier counter (zeroed by init)
                             float* partials)        // [2][NWG] double-buffered by t parity
{
  __shared__ float xbuf[2][ROWS_PER_WG][DPAD];       // double-buffered x tiles (async DMA dst)
  __shared__ float ac_l[DFEAT];                      // per-feature ac
  __shared__ float g_l[DFEAT];                       // sigmoid(tw)*sigmoid(ig) precomputed
  __shared__ float wred[ROWS_PER_WG];
  __shared__ float bshare;

  const int tid  = threadIdx.x;
  const int wid  = tid >> 5;                         // wave id = local row
  const int lane = tid & 31;
  const int row_base = blockIdx.x * ROWS_PER_WG;
  const int b = row_base + wid;

  // Issue one timestep's x tile via the async DMA engine. Uniform 20 issues per
  // wave (tail vectors clamped -> benign duplicate identical writes) so the
  // per-wave ASYNCcnt increment is a compile-time constant.
  auto issue_tile = [&](int t, int buf) {
    const float* base = x + ((size_t)t * BATCH + row_base) * DFEAT;
#pragma unroll
    for (int r = 0; r < ROWS_PER_WG; ++r) {
#pragma unroll
      for (int k = 0; k < PF_PER_THREAD; ++k) {
        int q = tid + k * THREADS;
        if (q > NVEC_ROW - 1) q = NVEC_ROW - 1;
        async_copy_b128(&xbuf[buf][r][q * 4], base + (size_t)r * DFEAT + q * 4);
      }
    }
  };

  issue_tile(0, 0);                                  // start DMA of t=0 immediately

  const float md = p_md[0];
  const float sd = p_sd[0];
  const float th = p_th[0];

  for (int idx = tid; idx < DFEAT; idx += THREADS) {
    ac_l[idx] = ac[idx];
    g_l[idx]  = sigmoidf_fast(tw[idx]) * sigmoidf_fast(ig[idx]);
  }

  float ist[KMAX], vst[KMAX];                        // LIF state lives in VGPRs for all T
#pragma unroll
  for (int k = 0; k < KMAX; ++k) { ist[k] = 0.0f; vst[k] = 0.0f; }
  __syncthreads();

#pragma unroll 1
  for (int t = 0; t < T_STEPS; ++t) {
    const int buf = t & 1;
    if (t + 1 < T_STEPS) {
      issue_tile(t + 1, buf ^ 1);                    // overlap next DMA with this compute
      s_wait_asynccnt_prev_tile();                   // in-order: older tile now complete
    } else {
      s_wait_asynccnt_zero();
    }
    __syncthreads();                                 // tile visible to all waves

    float* xl = &xbuf[buf][wid][0];

    // pass 1: row max (softmax stabilization)
    float m = -3.402823466e38f;
#pragma unroll
    for (int k = 0; k < KMAX; ++k) {
      int col = (k << 5) + lane;
      if (col < DFEAT) m = fmaxf(m, xl[col]);
    }
    m = wave_max(m);

    // pass 2: Z = sum e^(x-m), W = sum e^(x-m)*(x-m); entropy = logZ - W/Z
    float z = 0.0f, w = 0.0f;
#pragma unroll
    for (int k = 0; k < KMAX; ++k) {
      int col = (k << 5) + lane;
      if (col < DFEAT) {
        float d = xl[col] - m;
        float e = __expf(d);
        z += e;
        w = fmaf(e, d, w);
      }
    }
    z = wave_sum(z);
    w = wave_sum(w);
    const float ent = __logf(z) - w * __builtin_amdgcn_rcpf(z);

    // pass 3: pre-binding compressed; overwrite tile in LDS; accumulate |c|
    float asum = 0.0f;
#pragma unroll
    for (int k = 0; k < KMAX; ++k) {
      int col = (k << 5) + lane;
      if (col < DFEAT) {
        float c = xl[col] * sigmoidf_fast(ac_l[col] * ent) * g_l[col];
        xl[col] = c;
        asum += fabsf(c);
      }
    }
    asum = wave_sum(asum);
    if (lane == 0) wred[wid] = asum;
    __syncthreads();
    if (tid == 0) {
      float sblk = wred[0] + wred[1] + wred[2] + wred[3];
      __hip_atomic_store(&partials[buf * NWG + blockIdx.x], sblk,
                         __ATOMIC_RELEASE, __HIP_MEMORY_SCOPE_AGENT);
    }

    // grid-wide barrier (persistent kernel; deterministic fixed-order reduce after)
    __syncthreads();
    if (tid == 0) {
      __hip_atomic_fetch_add(ctl, 1u, __ATOMIC_ACQ_REL, __HIP_MEMORY_SCOPE_AGENT);
      const unsigned tgt = (unsigned)(t + 1) * (unsigned)NWG;
      while (__hip_atomic_load(ctl, __ATOMIC_ACQUIRE, __HIP_MEMORY_SCOPE_AGENT) < tgt)
        __builtin_amdgcn_s_sleep(2);
    }
    __syncthreads();

    // binding_strength = mean(|compressed|): reduce 256 partials in fixed order
    {
      float s0 = __hip_atomic_load(&partials[buf * NWG + tid],
                                   __ATOMIC_ACQUIRE, __HIP_MEMORY_SCOPE_AGENT);
      float s1 = __hip_atomic_load(&partials[buf * NWG + tid + THREADS],
                                   __ATOMIC_ACQUIRE, __HIP_MEMORY_SCOPE_AGENT);
      float s = wave_sum(s0 + s1);
      if (lane == 0) wred[wid] = s;
      __syncthreads();
      if (tid == 0)
        bshare = (wred[0] + wred[1] + wred[2] + wred[3]) *
                 (1.0f / ((float)BATCH * (float)DFEAT));
      __syncthreads();
    }
    const float binding = bshare;

    // pass 4: LIF recurrence (state in VGPRs) + surrogate spike out
    float* orow = out + ((size_t)t * BATCH + b) * DFEAT;
#pragma unroll
    for (int k = 0; k < KMAX; ++k) {
      int col = (k << 5) + lane;
      if (col < DFEAT) {
        float c   = xl[col] * binding;
        float cur = fmaf(sd, ist[k], c);
        ist[k] = cur;
        float vv  = fmaf(md, vst[k], cur);
        float s   = sigmoidf_fast(vv - th);
        vst[k] = vv - s * th;
        orow[col] = s;
      }
    }
    __syncthreads();   // all waves done with this tile before it is DMA-overwritten
  }
}

// ---- host entry ------------------------------------------------------------
extern "C" void kernel_launch(void* const* d_in, const int* in_sizes, int n_in,
                              void* d_out, int out_size, void* d_ws, size_t ws_size,
                              hipStream_t stream) {
  const float* x  = (const float*)d_in[0];
  const float* ac = (const float*)d_in[1];
  const float* tw = (const float*)d_in[2];
  const float* ig = (const float*)d_in[3];
  const float* md = (const float*)d_in[4];
  const float* sd = (const float*)d_in[5];
  const float* th = (const float*)d_in[6];
  float* out = (float*)d_out;

  unsigned* ctl   = (unsigned*)d_ws;                    // barrier counter (re-zeroed per call)
  float* partials = (float*)((char*)d_ws + 256);        // 2*NWG floats, fully rewritten

  init_ctl_kernel<<<dim3(1), dim3(32), 0, stream>>>(ctl);
  snn_atic_lif_persistent<<<dim3(NWG), dim3(THREADS), 0, stream>>>(
      x, ac, tw, ig, md, sd, th, out, ctl, partials);

  (void)in_sizes; (void)n_in; (void)out_size; (void)ws_size;
}